// DoRA_Linear_61813169324204
// MI455X (gfx1250) — compile-verified
//
#include <hip/hip_runtime.h>
#include <math.h>

#define IN_DIM  4096
#define OUT_DIM 4096
#define RANK    16
#define M_TOTAL 8192   // 4 * 2048
#define DORA_SCALE 2.0f

typedef __attribute__((ext_vector_type(2))) float v2f;
typedef __attribute__((ext_vector_type(8))) float v8f;

// ---------------------------------------------------------------------------
// Kernel A: w_v = W + DORA_SCALE * (B @ A)  -> workspace
//           norm_scale[o] = weight_m[o] / ||w_v[o,:]||_2
// One 256-thread block per output row.
// ---------------------------------------------------------------------------
__global__ __launch_bounds__(256) void dora_weight_prep(
    const float* __restrict__ W,      // [OUT, IN]
    const float* __restrict__ wm,     // [OUT]
    const float* __restrict__ A,      // [RANK, IN]
    const float* __restrict__ B,      // [OUT, RANK]
    float* __restrict__ wv,           // [OUT, IN] (workspace)
    float* __restrict__ nscale)       // [OUT]     (workspace)
{
    const int o   = blockIdx.x;
    const int tid = threadIdx.x;

    float brow[RANK];
#pragma unroll
    for (int r = 0; r < RANK; ++r)
        brow[r] = B[o * RANK + r] * DORA_SCALE;

    const float* wrow = W  + (size_t)o * IN_DIM;
    float*       vrow = wv + (size_t)o * IN_DIM;

    float ss = 0.0f;
    for (int i = tid; i < IN_DIM; i += 256) {
        float acc = wrow[i];
#pragma unroll
        for (int r = 0; r < RANK; ++r)
            acc += brow[r] * A[r * IN_DIM + i];
        vrow[i] = acc;
        ss += acc * acc;
    }

    // wave32 reduction, then cross-wave via LDS
#pragma unroll
    for (int off = 16; off > 0; off >>= 1)
        ss += __shfl_xor(ss, off, 32);

    __shared__ float red[8];
    if ((tid & 31) == 0) red[tid >> 5] = ss;
    __syncthreads();
    if (tid == 0) {
        float tot = 0.0f;
#pragma unroll
        for (int w = 0; w < 8; ++w) tot += red[w];
        nscale[o] = wm[o] / sqrtf(tot);
    }
}

// ---------------------------------------------------------------------------
// Kernel B: out[m,n] = nscale[n] * sum_k x[m,k]*wv[n,k] + bias[n]
// One wave (32 threads) per 64x64 output tile; 4x4 grid of 16x16 WMMA tiles,
// K stepped by 4 using V_WMMA_F32_16X16X4_F32.
//
// Fragment layouts (CDNA5 ISA 7.12.2):
//   A 16x4 f32 : lane&15 = M row; v0 = K(0 or 2), v1 = K(1 or 3) by lane half
//   B 4x16 f32 : lane&15 = N col; v0/v1 same K split  (B[k][n] = wv[n][k])
//   C/D 16x16  : VGPR g -> row g (lanes 0-15) / row g+8 (lanes 16-31), col = lane&15
// ---------------------------------------------------------------------------
__global__ __launch_bounds__(32) void dora_gemm_wmma(
    const float* __restrict__ x,       // [M_TOTAL, IN]
    const float* __restrict__ wv,      // [OUT, IN]
    const float* __restrict__ nscale,  // [OUT]
    const float* __restrict__ bias,    // [OUT]
    float* __restrict__ out)           // [M_TOTAL, OUT]
{
    const int lane  = threadIdx.x;   // 0..31
    const int lr    = lane & 15;
    const int kh    = lane >> 4;     // 0 or 1 -> K offset 0 / 2
    const int mBase = blockIdx.y * 64;
    const int nBase = blockIdx.x * 64;

    v8f acc[4][4];
#pragma unroll
    for (int i = 0; i < 4; ++i)
#pragma unroll
        for (int j = 0; j < 4; ++j)
            acc[i][j] = (v8f){0.f,0.f,0.f,0.f,0.f,0.f,0.f,0.f};

    const float* xp = x  + (size_t)(mBase + lr) * IN_DIM + kh * 2;
    const float* wp = wv + (size_t)(nBase + lr) * IN_DIM + kh * 2;

    for (int k = 0; k < IN_DIM; k += 4) {
        v2f a[4], b[4];
#pragma unroll
        for (int t = 0; t < 4; ++t) {
            a[t] = *(const v2f*)(xp + (size_t)t * 16 * IN_DIM);
            b[t] = *(const v2f*)(wp + (size_t)t * 16 * IN_DIM);
        }
#pragma unroll
        for (int i = 0; i < 4; ++i)
#pragma unroll
            for (int j = 0; j < 4; ++j)
                acc[i][j] = __builtin_amdgcn_wmma_f32_16x16x4_f32(
                    /*neg_a=*/false, a[i], /*neg_b=*/false, b[j],
                    /*c_mod=*/(short)0, acc[i][j],
                    /*reuse_a=*/false, /*reuse_b=*/false);
        xp += 4;
        wp += 4;
    }

    // Epilogue: fold in per-column norm scale and bias.
#pragma unroll
    for (int j = 0; j < 4; ++j) {
        const int col = nBase + j * 16 + lr;
        const float sc = nscale[col];
        const float bs = bias[col];
#pragma unroll
        for (int i = 0; i < 4; ++i) {
#pragma unroll
            for (int g = 0; g < 8; ++g) {
                const int row = mBase + i * 16 + g + kh * 8;
                out[(size_t)row * OUT_DIM + col] = acc[i][j][g] * sc + bs;
            }
        }
    }
}

// ---------------------------------------------------------------------------
extern "C" void kernel_launch(void* const* d_in, const int* in_sizes, int n_in,
                              void* d_out, int out_size, void* d_ws, size_t ws_size,
                              hipStream_t stream) {
    const float* x    = (const float*)d_in[0];  // [4,2048,4096]
    const float* W    = (const float*)d_in[1];  // [4096,4096]
    const float* bias = (const float*)d_in[2];  // [4096]
    const float* wm   = (const float*)d_in[3];  // [4096]
    const float* A    = (const float*)d_in[4];  // [16,4096]
    const float* B    = (const float*)d_in[5];  // [4096,16]
    float* out = (float*)d_out;

    float* wv     = (float*)d_ws;                       // 4096*4096 f32 = 64 MB
    float* nscale = wv + (size_t)OUT_DIM * IN_DIM;      // + 4096 f32

    dora_weight_prep<<<OUT_DIM, 256, 0, stream>>>(W, wm, A, B, wv, nscale);

    dim3 grid(OUT_DIM / 64, M_TOTAL / 64);  // (64, 128)
    dora_gemm_wmma<<<grid, 32, 0, stream>>>(x, wv, nscale, bias, out);
}